// Moe_layer_85718957294272
// MI455X (gfx1250) — compile-verified
//
#include <hip/hip_runtime.h>

typedef float v2f __attribute__((ext_vector_type(2)));
typedef float v8f __attribute__((ext_vector_type(8)));

#define DIN   1280
#define DOUT  1280
#define NSEQ  4096
#define NEXP  5
#define RLORA 16
#define KL    80      // NEXP * RLORA
#define BM    128
#define BN    128
#define KB    32      // K-chunk staged in LDS
#define LDP   36      // padded LDS row stride in floats (144B, 16B-aligned)

// D = A(16x4 f32) * B(4x16 f32) + C(16x16 f32)
static __device__ __forceinline__ v8f wmma_f32_16x16x4(v2f a, v2f b, v8f c) {
  return __builtin_amdgcn_wmma_f32_16x16x4_f32(false, a, false, b, (short)0, c,
                                               false, false);
}

// ---------------------------------------------------------------------------
// Kernel 1: h[m, e*16+r] = 0.5 * gates[b,e] * sum_d x[m,d] * lora_A[e,r,d]
// lora_A viewed flat as [KL=80, DIN] (contiguous). Grid: M/128 blocks, 256 thr.
// Each wave owns 16 rows x 80 cols => 5 fp32 16x16 accumulators.
// ---------------------------------------------------------------------------
__global__ __launch_bounds__(256) void moe_lora_h_kernel(
    const float* __restrict__ x, const float* __restrict__ loraA,
    const float* __restrict__ gates, float* __restrict__ hbuf) {
  __shared__ float sx[BM * LDP];
  __shared__ float sa[KL * LDP];
  const int tid  = threadIdx.x;
  const int lane = tid & 31;
  const int wave = tid >> 5;      // 0..7 -> 16 rows each
  const int lhi  = lane >> 4;     // 0/1
  const int llo  = lane & 15;
  const int mbase = blockIdx.x * BM;

  v8f acc[NEXP] = {};

  for (int kc = 0; kc < DIN; kc += KB) {
    __syncthreads();
#pragma unroll
    for (int i = 0; i < 4; ++i) {           // x tile: 128 x 32 floats
      int idx = tid + i * 256;              // 0..1023 float4 slots
      int r = idx >> 3, c = (idx & 7) * 4;
      *(float4*)(&sx[r * LDP + c]) =
          *(const float4*)(&x[(mbase + r) * DIN + kc + c]);
    }
    for (int idx = tid; idx < KL * 8; idx += 256) {  // A tile: 80 x 32 floats
      int r = idx >> 3, c = (idx & 7) * 4;
      *(float4*)(&sa[r * LDP + c]) =
          *(const float4*)(&loraA[r * DIN + kc + c]);
    }
    __syncthreads();
#pragma unroll
    for (int ks = 0; ks < KB; ks += 4) {
      v2f a = *(const v2f*)(&sx[(wave * 16 + llo) * LDP + ks + 2 * lhi]);
#pragma unroll
      for (int e = 0; e < NEXP; ++e) {
        v2f bf = *(const v2f*)(&sa[(e * 16 + llo) * LDP + ks + 2 * lhi]);
        acc[e] = wmma_f32_16x16x4(a, bf, acc[e]);
      }
    }
  }

  const int b = mbase / NSEQ;   // 128 | 4096, tiles never straddle batches
#pragma unroll
  for (int e = 0; e < NEXP; ++e) {
    float g = gates[b * NEXP + e] * 0.5f;   // lora scaling alpha/r = 0.5
#pragma unroll
    for (int v = 0; v < 8; ++v) {
      int m = mbase + wave * 16 + v + 8 * lhi;
      hbuf[m * KL + e * 16 + llo] = acc[e][v] * g;
    }
  }
}

// ---------------------------------------------------------------------------
// Kernel 2: out[m,o] = bias[o] + sum_d x[m,d]*W[o,d] + sum_k h[m,k]*Bc[o,k]
// where Bc[o, e*16+r] = lora_B[e,o,r]. Grid: (M/128, DOUT/128), 256 threads.
// 8 waves in a 4x2 (M x N) arrangement; each wave = 32x64 = 2x4 accumulators.
// Main K=1280 loop is LDS-staged; rank-80 LoRA tail uses direct global loads
// (400KB of lora_B + 2.6MB of h stay L2-resident).
// ---------------------------------------------------------------------------
__global__ __launch_bounds__(256) void moe_lora_out_kernel(
    const float* __restrict__ x, const float* __restrict__ W,
    const float* __restrict__ bias, const float* __restrict__ loraB,
    const float* __restrict__ hbuf, float* __restrict__ out) {
  __shared__ float sx[BM * LDP];
  __shared__ float sw[BN * LDP];
  const int tid  = threadIdx.x;
  const int lane = tid & 31;
  const int wave = tid >> 5;
  const int lhi  = lane >> 4;
  const int llo  = lane & 15;
  const int wmo  = (wave & 3) * 32;   // wave M-origin within tile
  const int wno  = (wave >> 2) * 64;  // wave N-origin within tile
  const int mbase = blockIdx.x * BM;
  const int nbase = blockIdx.y * BN;

  v8f acc[2][4] = {};

  for (int kc = 0; kc < DIN; kc += KB) {
    __syncthreads();
#pragma unroll
    for (int i = 0; i < 4; ++i) {
      int idx = tid + i * 256;
      int r = idx >> 3, c = (idx & 7) * 4;
      *(float4*)(&sx[r * LDP + c]) =
          *(const float4*)(&x[(mbase + r) * DIN + kc + c]);
      *(float4*)(&sw[r * LDP + c]) =
          *(const float4*)(&W[(nbase + r) * DIN + kc + c]);
      if (kc + KB < DIN) {  // pull next chunk toward the caches
        __builtin_prefetch(&x[(mbase + r) * DIN + kc + KB + c], 0, 1);
        __builtin_prefetch(&W[(nbase + r) * DIN + kc + KB + c], 0, 1);
      }
    }
    __syncthreads();
#pragma unroll
    for (int ks = 0; ks < KB; ks += 4) {
      v2f a[2], bf[4];
#pragma unroll
      for (int s = 0; s < 2; ++s)
        a[s] = *(const v2f*)(&sx[(wmo + s * 16 + llo) * LDP + ks + 2 * lhi]);
#pragma unroll
      for (int t = 0; t < 4; ++t)
        bf[t] = *(const v2f*)(&sw[(wno + t * 16 + llo) * LDP + ks + 2 * lhi]);
#pragma unroll
      for (int s = 0; s < 2; ++s)
#pragma unroll
        for (int t = 0; t < 4; ++t)
          acc[s][t] = wmma_f32_16x16x4(a[s], bf[t], acc[s][t]);
    }
  }

  // LoRA tail: K = 80 (gates/scaling already folded into hbuf)
  for (int kb = 0; kb < KL; kb += 4) {
    int k0 = kb + 2 * lhi;            // even, e-block never straddled
    int e = k0 >> 4, r = k0 & 15;
    v2f a[2], bf[4];
#pragma unroll
    for (int s = 0; s < 2; ++s)
      a[s] = *(const v2f*)(&hbuf[(mbase + wmo + s * 16 + llo) * KL + k0]);
#pragma unroll
    for (int t = 0; t < 4; ++t) {
      int o = nbase + wno + t * 16 + llo;
      bf[t] = *(const v2f*)(&loraB[(e * DOUT + o) * RLORA + r]);
    }
#pragma unroll
    for (int s = 0; s < 2; ++s)
#pragma unroll
      for (int t = 0; t < 4; ++t)
        acc[s][t] = wmma_f32_16x16x4(a[s], bf[t], acc[s][t]);
  }

  // Epilogue: + bias, store. C/D layout: M = v + 8*lhi, N = llo.
#pragma unroll
  for (int s = 0; s < 2; ++s) {
#pragma unroll
    for (int t = 0; t < 4; ++t) {
      int o = nbase + wno + t * 16 + llo;
      float bv = bias[o];
#pragma unroll
      for (int v = 0; v < 8; ++v) {
        int m = mbase + wmo + s * 16 + v + 8 * lhi;
        out[m * DOUT + o] = acc[s][t][v] + bv;
      }
    }
  }
}

extern "C" void kernel_launch(void* const* d_in, const int* in_sizes, int n_in,
                              void* d_out, int out_size, void* d_ws,
                              size_t ws_size, hipStream_t stream) {
  (void)in_sizes; (void)n_in; (void)out_size; (void)ws_size;
  const float* x     = (const float*)d_in[0];  // [2,4096,1280]
  const float* gates = (const float*)d_in[1];  // [2,5]
  const float* W     = (const float*)d_in[2];  // [1280,1280]
  const float* bias  = (const float*)d_in[3];  // [1280]
  const float* loraA = (const float*)d_in[4];  // [5,16,1280] == [80,1280]
  const float* loraB = (const float*)d_in[5];  // [5,1280,16]
  float* out  = (float*)d_out;                 // [2,4096,1280]
  float* hbuf = (float*)d_ws;                  // [8192,80] = 2.62 MB scratch

  const int M = 2 * NSEQ;                      // 8192 rows
  moe_lora_h_kernel<<<dim3(M / BM), dim3(256), 0, stream>>>(x, loraA, gates,
                                                            hbuf);
  moe_lora_out_kernel<<<dim3(M / BM, DOUT / BN), dim3(256), 0, stream>>>(
      x, W, bias, loraB, hbuf, out);
}